// CaptionModel_9156870275335
// MI455X (gfx1250) — compile-verified
//
#include <hip/hip_runtime.h>
#include <math.h>

#define VOCAB 32000
#define HID   1024
#define BEAM  16
#define TSTEPS 16
#define NTILES (VOCAB / 16)

typedef __attribute__((ext_vector_type(2))) float v2f;
typedef __attribute__((ext_vector_type(8))) float v8f;
typedef __attribute__((ext_vector_type(4))) unsigned int u32x4;
typedef __attribute__((ext_vector_type(4))) int i32x4;
typedef __attribute__((ext_vector_type(8))) int i32x8;

__device__ __forceinline__ v8f wmma_f32x4(v2f a, v2f b, v8f c) {
  // D = A(16x4) * B(4x16) + C(16x16), f32. Confirmed lowering:
  // v_wmma_f32_16x16x4_f32 v[d:d+7], v[a:a+1], v[b:b+1], v[c:c+7]
  return __builtin_amdgcn_wmma_f32_16x16x4_f32(false, a, false, b, (short)0, c,
                                               false, false);
}

// ---------------------------------------------------------------- init
__global__ void k0_init(const float* __restrict__ state, float* hA,
                        int* beam_seq, float* beam_seq_lp, float* beam_lps,
                        float* Z) {
  int idx = blockIdx.x * blockDim.x + threadIdx.x;
  if (idx < BEAM * HID) hA[idx] = state[idx];
  if (idx < TSTEPS * BEAM) { beam_seq[idx] = 0; beam_seq_lp[idx] = 0.f; }
  if (idx < BEAM) { beam_lps[idx] = 0.f; Z[idx] = 0.f; }
}

// ------------------------------------------------- top-16 per beam (LDS)
// One block per beam. The 128 KB beam row is staged into LDS by the Tensor
// Data Mover (1-D D# descriptor, TENSORcnt-tracked), then normalized
// (x - Z[beam], -1000 penalty on the last vocab entry) and scanned with 16
// argmax sweeps, knocking each winner out to -inf.
__global__ void k1_topk(const float* __restrict__ logits,
                        const float* __restrict__ Z, float* topv, int* topi) {
  extern __shared__ float xs[];  // VOCAB floats (dynamic LDS)
  __shared__ float sval[512];
  __shared__ int sidx[512];
  const int beam = blockIdx.x, tid = threadIdx.x, nt = blockDim.x;
  const float* row = logits + (size_t)beam * VOCAB;

#if __has_builtin(__builtin_amdgcn_tensor_load_to_lds)
  if (tid < 32) {  // wave 0 issues the DMA; TDM ignores EXEC
    const unsigned int lds_off = (unsigned int)(uintptr_t)(void*)xs;
    const unsigned long long ga = (unsigned long long)(uintptr_t)row;
    u32x4 g0;
    g0.x = 1u;                                   // count=1, user descriptor
    g0.y = lds_off;                              // bits 63:32  lds_addr
    g0.z = (unsigned int)(ga & 0xffffffffu);     // bits 95:64  global_addr lo
    g0.w = (unsigned int)((ga >> 32) & 0x1ffffffu) | 0x80000000u;  // type=2
    i32x8 g1;
    g1[0] = 0x20000;            // data_size=2 (4B), workgroup_mask=0
    g1[1] = (int)((VOCAB & 0xffffu) << 16);      // tensor_dim0 lo16
    g1[2] = (int)((VOCAB >> 16) & 0xffffu);      // tensor_dim0 hi16, dim1=0
    g1[3] = (int)((VOCAB & 0xffffu) << 16);      // tile_dim0 = VOCAB
    g1[4] = 0;                                   // tile_dim1/2 unused
    g1[5] = VOCAB;                               // tensor_dim0_stride lo32
    g1[6] = 0;
    g1[7] = 0;
    i32x4 gz = {0, 0, 0, 0};
#if __clang_major__ >= 23
    i32x8 gz8 = {0, 0, 0, 0, 0, 0, 0, 0};
    __builtin_amdgcn_tensor_load_to_lds(g0, g1, gz, gz, gz8, 0);
#else
    __builtin_amdgcn_tensor_load_to_lds(g0, g1, gz, gz, 0);
#endif
#if __has_builtin(__builtin_amdgcn_s_wait_tensorcnt)
    __builtin_amdgcn_s_wait_tensorcnt(0);
#endif
  }
  __syncthreads();
#else
  for (int j = tid; j < VOCAB; j += nt) xs[j] = row[j];
  __syncthreads();
#endif

  // normalize in LDS: lpf = logits - Z  (+ last-token penalty)
  const float zb = Z[beam];
  for (int j = tid; j < VOCAB; j += nt) {
    float x = xs[j] - zb;
    if (j == VOCAB - 1) x -= 1000.f;
    xs[j] = x;
  }
  __syncthreads();

  for (int k = 0; k < BEAM; ++k) {
    float bv = -INFINITY; int bi = VOCAB;
    for (int j = tid; j < VOCAB; j += nt) {
      float x = xs[j];
      if (x > bv || (x == bv && j < bi)) { bv = x; bi = j; }
    }
    sval[tid] = bv; sidx[tid] = bi;
    __syncthreads();
    for (int s = nt >> 1; s > 0; s >>= 1) {
      if (tid < s) {
        float ov = sval[tid + s]; int oi = sidx[tid + s];
        if (ov > sval[tid] || (ov == sval[tid] && oi < sidx[tid])) {
          sval[tid] = ov; sidx[tid] = oi;
        }
      }
      __syncthreads();
    }
    if (tid == 0) {
      topv[beam * BEAM + k] = sval[0];
      topi[beam * BEAM + k] = sidx[0];
      xs[sidx[0]] = -INFINITY;   // knock out winner
    }
    __syncthreads();
  }
}

// ------------------------------------- global top-16 + gather/permute
__global__ void k2_combine(const float* __restrict__ topv,
                           const int* __restrict__ topi, float* beam_lps,
                           int* cbuf, int* qbuf, int* beam_seq,
                           float* beam_seq_lp, int t) {
  __shared__ float cand[BEAM * BEAM];
  __shared__ int oldseq[TSTEPS * BEAM];
  __shared__ float oldlp[TSTEPS * BEAM];
  __shared__ int selq[BEAM], selc[BEAM];
  __shared__ float selr[BEAM], selp[BEAM];
  const int tid = threadIdx.x;          // 256 threads
  const int q = tid >> 4, col = tid & 15;
  float v = beam_lps[q] + topv[tid];
  if (t == 0 && q > 0) v = -INFINITY;   // row_ok mask at t==0
  cand[tid] = v;
  oldseq[tid] = beam_seq[tid];
  oldlp[tid] = beam_seq_lp[tid];
  __syncthreads();
  if (tid == 0) {
    for (int k = 0; k < BEAM; ++k) {
      float bv = cand[0]; int bi = 0;
      for (int i = 1; i < BEAM * BEAM; ++i)
        if (cand[i] > bv) { bv = cand[i]; bi = i; }
      selp[k] = bv; selq[k] = bi >> 4;
      selc[k] = topi[bi]; selr[k] = topv[bi];
      cand[bi] = -INFINITY;
    }
  }
  __syncthreads();
  // beam_seq = beam_seq[:, q]; beam_seq[t] = c   (same for lp / r)
  if (q == t) { beam_seq[tid] = selc[col]; beam_seq_lp[tid] = selr[col]; }
  else {
    beam_seq[tid]    = oldseq[q * BEAM + selq[col]];
    beam_seq_lp[tid] = oldlp[q * BEAM + selq[col]];
  }
  if (tid < BEAM) {
    const bool done = (selc[tid] == 0) || (t == TSTEPS - 1);
    beam_lps[tid] = done ? -1000.f : selp[tid];
    cbuf[tid] = selc[tid];
    qbuf[tid] = selq[tid];
  }
}

// ------------------- h = tanh(embed[c] @ W_ih + h_prev[q] @ W_hh + b)
__global__ void __launch_bounds__(32)
k3_rnn(const float* __restrict__ embed, const float* __restrict__ Wih,
       const float* __restrict__ Whh, const float* __restrict__ bias,
       const float* __restrict__ hprev, const int* __restrict__ cbuf,
       const int* __restrict__ qbuf, float* __restrict__ hcur) {
  const int lane = threadIdx.x;
  const int n0 = blockIdx.x * 16;
  const int m = lane & 15;            // A row for this lane
  const int n = lane & 15;            // B/C column for this lane
  const int kb = (lane >> 4) * 2;     // K sub-offset (0 or 2)
  const float* A1 = embed + (size_t)cbuf[m] * HID;
  const float* A2 = hprev + (size_t)qbuf[m] * HID;
  v8f acc = {0.f, 0.f, 0.f, 0.f, 0.f, 0.f, 0.f, 0.f};
  for (int k0 = 0; k0 < HID; k0 += 4) {
    v2f a1; a1.x = A1[k0 + kb]; a1.y = A1[k0 + kb + 1];
    v2f b1; b1.x = Wih[(size_t)(k0 + kb) * HID + n0 + n];
            b1.y = Wih[(size_t)(k0 + kb + 1) * HID + n0 + n];
    acc = wmma_f32x4(a1, b1, acc);
    v2f a2; a2.x = A2[k0 + kb]; a2.y = A2[k0 + kb + 1];
    v2f b2; b2.x = Whh[(size_t)(k0 + kb) * HID + n0 + n];
            b2.y = Whh[(size_t)(k0 + kb + 1) * HID + n0 + n];
    acc = wmma_f32x4(a2, b2, acc);
  }
  const float bn = bias[n0 + n];
  const int rbase = (lane >> 4) * 8;  // C layout: VGPR r -> M=r (+8 hi half)
  for (int r = 0; r < 8; ++r)
    hcur[(size_t)(rbase + r) * HID + n0 + n] = tanhf(acc[r] + bn);
}

// ------------- logits = h @ W_out  (two tiles/wave, dual accumulators)
// Two independent wmma chains per wave: breaks the accumulator RAW chain,
// shares A fragments, and each B row access covers a full 128B cacheline.
__global__ void __launch_bounds__(128)
k4a_logits(const float* __restrict__ h, const float* __restrict__ Wout,
           float* __restrict__ logits, float* __restrict__ pm,
           float* __restrict__ ps) {
  const int lane = threadIdx.x & 31;
  const int job = blockIdx.x * 4 + (threadIdx.x >> 5);  // 0..999
  const int n0 = job * 32;
  const int m = lane & 15, n = lane & 15;
  const int kb = (lane >> 4) * 2;
  const float* hr = h + (size_t)m * HID;
  v8f acc0 = {0.f, 0.f, 0.f, 0.f, 0.f, 0.f, 0.f, 0.f};
  v8f acc1 = {0.f, 0.f, 0.f, 0.f, 0.f, 0.f, 0.f, 0.f};
  for (int k0 = 0; k0 < HID; k0 += 4) {
    v2f a; a.x = hr[k0 + kb]; a.y = hr[k0 + kb + 1];
    const float* w0 = Wout + (size_t)(k0 + kb) * VOCAB + n0 + n;
    const float* w1 = w0 + VOCAB;                 // next K row
    v2f b0; b0.x = w0[0];  b0.y = w1[0];
    v2f b1; b1.x = w0[16]; b1.y = w1[16];
    if (k0 + 64 < HID)  // keep the W_out stream ahead of the WMMAs
      __builtin_prefetch(w0 + (size_t)64 * VOCAB, 0, 1);
    acc0 = wmma_f32x4(a, b0, acc0);
    acc1 = wmma_f32x4(a, b1, acc1);
  }
  const int rbase = (lane >> 4) * 8;
  for (int half = 0; half < 2; ++half) {
    const v8f* accp = half ? &acc1 : &acc0;
    const int tile = job * 2 + half;
    const int c0 = n0 + half * 16;
    for (int r = 0; r < 8; ++r) {
      const int row = rbase + r;
      const float v = (*accp)[r];
      logits[(size_t)row * VOCAB + c0 + n] = v;
      float mx = v;                                  // row max within tile
      for (int off = 8; off >= 1; off >>= 1)
        mx = fmaxf(mx, __shfl_xor(mx, off, 16));
      float e = expf(v - mx);                        // sumexp rel. tile max
      for (int off = 8; off >= 1; off >>= 1) e += __shfl_xor(e, off, 16);
      if (n == 0) {
        pm[(size_t)row * NTILES + tile] = mx;
        ps[(size_t)row * NTILES + tile] = e;
      }
    }
  }
}

// ------------------- Z[beam] = logsumexp over all tile partials
__global__ void k4b_reduceZ(const float* __restrict__ pm,
                            const float* __restrict__ ps, float* Z) {
  __shared__ float sm[256], ss[256];
  const int beam = blockIdx.x, tid = threadIdx.x;
  float M = -INFINITY, S = 0.f;
  for (int i = tid; i < NTILES; i += 256) {
    const float m = pm[(size_t)beam * NTILES + i];
    const float s = ps[(size_t)beam * NTILES + i];
    if (m > M) { S = S * expf(M - m) + s; M = m; }
    else       { S += s * expf(m - M); }
  }
  sm[tid] = M; ss[tid] = S;
  __syncthreads();
  for (int sh = 128; sh > 0; sh >>= 1) {
    if (tid < sh) {
      float m2 = sm[tid + sh], s2 = ss[tid + sh], M1 = sm[tid], S1 = ss[tid];
      if (m2 > M1) { S1 = S1 * expf(M1 - m2) + s2; M1 = m2; }
      else         { S1 += s2 * expf(m2 - M1); }
      sm[tid] = M1; ss[tid] = S1;
    }
    __syncthreads();
  }
  if (tid == 0) Z[beam] = sm[0] + logf(ss[0]);
}

// ------------------- pack the three outputs into d_out
__global__ void k5_out(const int* __restrict__ beam_seq,
                       const float* __restrict__ beam_seq_lp,
                       const float* __restrict__ beam_lps, float* out) {
  const int tid = blockIdx.x * blockDim.x + threadIdx.x;
  if (tid < 256) ((int*)out)[tid] = beam_seq[tid];
  else if (tid < 512) out[tid] = beam_seq_lp[tid - 256];
  else if (tid < 528) out[tid] = beam_lps[tid - 512];
}

// ---------------------------------------------------------------- host
extern "C" void kernel_launch(void* const* d_in, const int* in_sizes, int n_in,
                              void* d_out, int out_size, void* d_ws,
                              size_t ws_size, hipStream_t stream) {
  const float* state    = (const float*)d_in[0];
  const float* logprobs = (const float*)d_in[1];
  const float* embed    = (const float*)d_in[2];
  const float* Wih      = (const float*)d_in[3];
  const float* Whh      = (const float*)d_in[4];
  const float* bias     = (const float*)d_in[5];
  const float* Wout     = (const float*)d_in[6];

  float* ws     = (float*)d_ws;
  float* logits = ws;                                  // 16*32000
  float* pm     = logits + BEAM * VOCAB;               // 16*2000
  float* ps     = pm + BEAM * NTILES;                  // 16*2000
  float* Z      = ps + BEAM * NTILES;                  // 16
  float* topv   = Z + BEAM;                            // 256
  int*   topi   = (int*)(topv + BEAM * BEAM);          // 256
  int*   cbuf   = topi + BEAM * BEAM;                  // 16
  int*   qbuf   = cbuf + BEAM;                         // 16
  float* blps   = (float*)(qbuf + BEAM);               // 16
  int*   bseq   = (int*)(blps + BEAM);                 // 256
  float* bslp   = (float*)(bseq + TSTEPS * BEAM);      // 256
  float* hA     = bslp + TSTEPS * BEAM;                // 16*1024
  float* hB     = hA + BEAM * HID;                     // 16*1024

  k0_init<<<64, 256, 0, stream>>>(state, hA, bseq, bslp, blps, Z);

  for (int t = 0; t < TSTEPS; ++t) {
    const float* lsrc = (t == 0) ? logprobs : logits;
    k1_topk<<<BEAM, 512, VOCAB * sizeof(float), stream>>>(lsrc, Z, topv, topi);
    k2_combine<<<1, 256, 0, stream>>>(topv, topi, blps, cbuf, qbuf, bseq, bslp, t);
    if (t < TSTEPS - 1) {
      float* hprev = (t & 1) ? hB : hA;
      float* hcur  = (t & 1) ? hA : hB;
      k3_rnn<<<HID / 16, 32, 0, stream>>>(embed, Wih, Whh, bias, hprev, cbuf,
                                          qbuf, hcur);
      k4a_logits<<<NTILES / 8, 128, 0, stream>>>(hcur, Wout, logits, pm, ps);
      k4b_reduceZ<<<BEAM, 256, 0, stream>>>(pm, ps, Z);
    }
  }
  k5_out<<<3, 256, 0, stream>>>(bseq, bslp, blps, (float*)d_out);
}